// Model_22969485099828
// MI455X (gfx1250) — compile-verified
//
#include <hip/hip_runtime.h>
#include <math.h>

typedef float v8f __attribute__((ext_vector_type(8)));
typedef float v2f __attribute__((ext_vector_type(2)));

#define NB   64
#define CCH  128
#define TT   128
#define VV   25
#define HH   8
#define CSUB 16

// ws layout (floats)
#define WS_W1P   0        // 3*8*32*32 = 24576
#define WS_BIAS  24576    // 128
#define WS_MEAN  24704    // 128
#define WS_ISTD  24832    // 128

// ---------------------------------------------------------------------------
// Kernel 1: build padded effective weights  w1_eff = fc1/||fc1||_col + pe/||pe||_col
// stored as [i][h][v(32)][w(32)], zero padded so WMMA tiles need no bounds checks.
// Also bias_sum[ch] = sum_i fc2_b[i][ch].
// ---------------------------------------------------------------------------
__global__ void prep_kernel(const float* __restrict__ fc1,
                            const float* __restrict__ rpe,
                            const int*   __restrict__ hops,
                            const float* __restrict__ fc2_b,
                            float* __restrict__ ws)
{
    int g = blockIdx.x * blockDim.x + threadIdx.x;
    if (g < 3 * HH * 32) {
        int ih = g >> 5;          // i*8 + h
        int wc = g & 31;          // output column
        float* W1 = ws + WS_W1P + ih * 32 * 32;
        if (wc >= VV) {
            for (int v = 0; v < 32; ++v) W1[v * 32 + wc] = 0.f;
        } else {
            const float* f = fc1 + ih * VV * VV;   // [v][w]
            const float* r = rpe + ih * VV;        // [V]
            float sf = 0.f, sp = 0.f;
            float pe[VV];
            for (int v = 0; v < VV; ++v) {
                float fv = f[v * VV + wc];
                sf += fv * fv;
                float pv = r[hops[v * VV + wc]];
                pe[v] = pv;
                sp += pv * pv;
            }
            float nf = sqrtf(sf) + 1e-4f;
            float np = sqrtf(sp) + 1e-4f;
            for (int v = 0; v < VV; ++v)
                W1[v * 32 + wc] = f[v * VV + wc] / nf + pe[v] / np;
            for (int v = VV; v < 32; ++v)
                W1[v * 32 + wc] = 0.f;
        }
    } else if (g < 3 * HH * 32 + CCH) {
        int ch = g - 3 * HH * 32;
        ws[WS_BIAS + ch] = fc2_b[ch] + fc2_b[CCH + ch] + fc2_b[2 * CCH + ch];
    }
}

// ---------------------------------------------------------------------------
// Kernel 2: fused  z = X*W1  (per head, over V) ; Y += WG*z (over c) ; sum i
// One workgroup per (n,h).  16 waves: wave = (t-block pair, w-half).
// Stage-1 results staged in LDS  z1[tblk][c][t][w]  (w padded to 33).
// Y accumulated in VGPRs across the 3 branches, bias added, written to y.
// ---------------------------------------------------------------------------
#define Z1_WPAD 33
#define Z1_FLOATS (8 * 16 * 16 * Z1_WPAD)

__global__ __launch_bounds__(512) void fused_gemm_kernel(
    const float* __restrict__ x,
    const float* __restrict__ fc2_w,
    const float* __restrict__ ws,
    float* __restrict__ y)
{
    extern __shared__ float z1[];          // [8][16][16][33]

    const int n    = blockIdx.x >> 3;
    const int h    = blockIdx.x & 7;
    const int lane = threadIdx.x & 31;
    const int wave = threadIdx.x >> 5;     // 0..15
    const int pair = wave >> 1;            // t-block 0..7
    const int half = wave & 1;             // c-half / w-half
    const int t0   = pair * 16;
    const int wl   = lane & 15;
    const int hi   = lane >> 4;            // 0 or 1
    const int kb   = hi << 1;              // K sub-offset for A/B frags
    const int wbase = half * 13;           // w-columns: 0..12 / 13..24(+pad 25)

    v8f Yacc[13] = {};                     // D frags, persist over i-loop

    for (int i = 0; i < 3; ++i) {
        // ---- B fragments of W1 (padded, unconditional) ----
        const float* W1 = ws + WS_W1P + (size_t)(i * HH + h) * 32 * 32;
        v2f bw[7][2];
#pragma unroll
        for (int ks = 0; ks < 7; ++ks) {
            int k = ks * 4 + kb;
#pragma unroll
            for (int wt = 0; wt < 2; ++wt) {
                bw[ks][wt].x = W1[(k    ) * 32 + wt * 16 + wl];
                bw[ks][wt].y = W1[(k + 1) * 32 + wt * 16 + wl];
            }
        }

        // ---- stage 1: z1[c][t][w] = X[c][t][:] * W1  for this wave's 8 c's ----
#pragma unroll 1
        for (int cc = 0; cc < 8; ++cc) {
            int c = half * 8 + cc;
            const float* Xr = x + (size_t)((n * CCH + h * CSUB + c) * TT + t0 + wl) * VV;
            v8f a0 = {}, a1 = {};
#pragma unroll
            for (int ks = 0; ks < 7; ++ks) {
                int v0 = ks * 4 + kb;
                v2f af;
                af.x = (v0     < VV) ? Xr[v0]     : 0.f;
                af.y = (v0 + 1 < VV) ? Xr[v0 + 1] : 0.f;
                a0 = __builtin_amdgcn_wmma_f32_16x16x4_f32(false, af, false, bw[ks][0],
                                                           (short)0, a0, false, false);
                a1 = __builtin_amdgcn_wmma_f32_16x16x4_f32(false, af, false, bw[ks][1],
                                                           (short)0, a1, false, false);
            }
            // scatter D frags to LDS: element (row m = j+8*hi = t_local, col = w)
            float* zb = z1 + (size_t)((pair * 16 + c) * 16) * Z1_WPAD;
#pragma unroll
            for (int j = 0; j < 8; ++j) {
                int m = j + 8 * hi;
                zb[m * Z1_WPAD + wl]      = a0[j];
                zb[m * Z1_WPAD + 16 + wl] = a1[j];
            }
        }
        __syncthreads();

        // ---- stage 2: Y[d][t] += WG[d][c] * z1[c][t][w]  (K = c = 16) ----
        const float* WG = fc2_w + (size_t)(i * CCH + h * CSUB) * CSUB;  // [d][c]
        v2f ag[4];
#pragma unroll
        for (int ks = 0; ks < 4; ++ks) {
            int k = ks * 4 + kb;
            ag[ks].x = WG[wl * CSUB + k];
            ag[ks].y = WG[wl * CSUB + k + 1];
        }
        const float* zt = z1 + (size_t)(pair * 16) * 16 * Z1_WPAD;
#pragma unroll
        for (int wi = 0; wi < 13; ++wi) {
            int w = wbase + wi;            // half==1,wi==12 -> w=25 (zero pad col)
            v8f acc = Yacc[wi];
#pragma unroll
            for (int ks = 0; ks < 4; ++ks) {
                int kc = ks * 4 + kb;
                v2f bf;
                bf.x = zt[(kc    ) * 16 * Z1_WPAD + wl * Z1_WPAD + w];
                bf.y = zt[(kc + 1) * 16 * Z1_WPAD + wl * Z1_WPAD + w];
                acc = __builtin_amdgcn_wmma_f32_16x16x4_f32(false, ag[ks], false, bf,
                                                            (short)0, acc, false, false);
            }
            Yacc[wi] = acc;
        }
        __syncthreads();
    }

    // ---- epilogue: add summed bias, write pre-BN y into d_out ----
    const float* bs = ws + WS_BIAS + h * CSUB;
#pragma unroll
    for (int wi = 0; wi < 13; ++wi) {
        int w = wbase + wi;
        if (w < VV) {
#pragma unroll
            for (int j = 0; j < 8; ++j) {
                int d = j + 8 * hi;
                float val = Yacc[wi][j] + bs[d];
                y[(size_t)((n * CCH + h * CSUB + d) * TT + t0 + wl) * VV + w] = val;
            }
        }
    }
}

// ---------------------------------------------------------------------------
// Kernel 3: per-channel mean / inv-std over (N,T,V).  Deterministic fixed-order
// reduction (no float atomics), double accumulation.
// ---------------------------------------------------------------------------
__global__ __launch_bounds__(256) void stats_kernel(const float* __restrict__ y,
                                                    float* __restrict__ ws)
{
    __shared__ double sred[256];
    __shared__ double s2red[256];
    const int ch  = blockIdx.x;
    const int tid = threadIdx.x;
    double s = 0.0, s2 = 0.0;
    for (int n = 0; n < NB; ++n) {
        const float* p = y + (size_t)(n * CCH + ch) * (TT * VV);
        for (int r = tid; r < TT * VV; r += 256) {
            double v = (double)p[r];
            s += v; s2 += v * v;
        }
    }
    sred[tid] = s; s2red[tid] = s2;
    __syncthreads();
    for (int off = 128; off > 0; off >>= 1) {
        if (tid < off) { sred[tid] += sred[tid + off]; s2red[tid] += s2red[tid + off]; }
        __syncthreads();
    }
    if (tid == 0) {
        double cnt  = (double)NB * TT * VV;
        double mean = sred[0] / cnt;
        double var  = s2red[0] / cnt - mean * mean;
        ws[WS_MEAN + ch] = (float)mean;
        ws[WS_ISTD + ch] = (float)(1.0 / sqrt(var + 1e-5));
    }
}

// ---------------------------------------------------------------------------
// Kernel 4: out = relu(gamma*(y-mean)*invstd + beta + x), in place on d_out.
// ---------------------------------------------------------------------------
__global__ void bn_relu_kernel(const float* __restrict__ x,
                               const float* __restrict__ gamma,
                               const float* __restrict__ beta,
                               const float* __restrict__ ws,
                               float* __restrict__ y,
                               size_t total)
{
    size_t idx    = (size_t)blockIdx.x * blockDim.x + threadIdx.x;
    size_t stride = (size_t)gridDim.x * blockDim.x;
    for (; idx < total; idx += stride) {
        int ch  = (int)((idx / (TT * VV)) % CCH);
        float v = y[idx];
        float r = gamma[ch] * (v - ws[WS_MEAN + ch]) * ws[WS_ISTD + ch]
                + beta[ch] + x[idx];
        y[idx] = r > 0.f ? r : 0.f;
    }
}

// ---------------------------------------------------------------------------
extern "C" void kernel_launch(void* const* d_in, const int* in_sizes, int n_in,
                              void* d_out, int out_size, void* d_ws, size_t ws_size,
                              hipStream_t stream)
{
    const float* x     = (const float*)d_in[0];
    const float* fc1   = (const float*)d_in[1];
    const float* rpe   = (const float*)d_in[2];
    const int*   hops  = (const int*)  d_in[3];
    const float* fc2w  = (const float*)d_in[4];
    const float* fc2b  = (const float*)d_in[5];
    const float* gamma = (const float*)d_in[6];
    const float* beta  = (const float*)d_in[7];
    float* out = (float*)d_out;
    float* ws  = (float*)d_ws;

    prep_kernel<<<4, 256, 0, stream>>>(fc1, rpe, hops, fc2b, ws);

    size_t lds_bytes = (size_t)Z1_FLOATS * sizeof(float);   // 270336 B < 320 KB
    fused_gemm_kernel<<<NB * HH, 512, lds_bytes, stream>>>(x, fc2w, ws, out);

    stats_kernel<<<CCH, 256, 0, stream>>>(out, ws);

    size_t total = (size_t)NB * CCH * TT * VV;
    bn_relu_kernel<<<2048, 256, 0, stream>>>(x, gamma, beta, ws, out, total);
}